// CrossCompressUnit_9019431321917
// MI455X (gfx1250) — compile-verified
//
#include <hip/hip_runtime.h>

typedef float v2f __attribute__((ext_vector_type(2)));
typedef float v8f __attribute__((ext_vector_type(8)));

#define DIM 256
#define ROWS_PER_WAVE 16
#define WAVES_PER_BLOCK 4
#define ROWS_PER_BLOCK (ROWS_PER_WAVE * WAVES_PER_BLOCK)      // 64
#define LDS_STRIDE 260                                        // 260 % 64 == 4 -> bank-conflict-free A reads
#define TILE_FLOATS (ROWS_PER_WAVE * LDS_STRIDE)              // 4160
#define WAVE_SLICE (2 * TILE_FLOATS + 64)                     // v-tile + e-tile + 16x4 scalars = 8384
#define WLDS_FLOATS (4 * DIM)                                 // 4 weight vectors
#define LDS_TOTAL_FLOATS (WLDS_FLOATS + WAVES_PER_BLOCK * WAVE_SLICE)  // 34560 floats = 138240 B

extern "C" __global__ __launch_bounds__(128)
void cross_compress_kernel(const float* __restrict__ v,
                           const float* __restrict__ e,
                           const float* __restrict__ w_vv,
                           const float* __restrict__ w_ev,
                           const float* __restrict__ w_ve,
                           const float* __restrict__ w_ee,
                           const float* __restrict__ bias_v,
                           const float* __restrict__ bias_e,
                           float* __restrict__ out_v,
                           float* __restrict__ out_e)
{
    extern __shared__ float smem[];
    float* wlds = smem;                                    // [4][256]: w_ev, w_ee, w_vv, w_ve

    const int tid      = threadIdx.x;
    const int wave     = tid >> 5;
    const int lane     = tid & 31;
    const int lane16   = lane & 15;
    const int laneHalf = lane >> 4;

    float* vbuf = smem + WLDS_FLOATS + wave * WAVE_SLICE;
    float* ebuf = vbuf + TILE_FLOATS;
    float* sbuf = ebuf + TILE_FLOATS;                      // [16 rows][4]: {v.w_ev, v.w_ee, e.w_vv, e.w_ve}

    // ---- load weight vectors into LDS (shared by block) ----
    for (int i = tid; i < 4 * DIM; i += 128) {
        const int which = i >> 8;
        const int c     = i & (DIM - 1);
        float w;
        if      (which == 0) w = w_ev[c];
        else if (which == 1) w = w_ee[c];
        else if (which == 2) w = w_vv[c];
        else                 w = w_ve[c];
        wlds[i] = w;
    }

    const long rowBase = (long)blockIdx.x * ROWS_PER_BLOCK + wave * ROWS_PER_WAVE;

    // ---- stage 16-row tiles of v and e, fully coalesced b128 ----
    for (int i = lane; i < ROWS_PER_WAVE * (DIM / 4); i += 32) {
        const int row = i >> 6;                            // DIM/4 == 64 chunks per row
        const int c4  = i & 63;
        const float4 vv = *(const float4*)(v + (rowBase + row) * DIM + c4 * 4);
        const float4 ee = *(const float4*)(e + (rowBase + row) * DIM + c4 * 4);
        *(float4*)(vbuf + row * LDS_STRIDE + c4 * 4) = vv;
        *(float4*)(ebuf + row * LDS_STRIDE + c4 * 4) = ee;
    }
    __syncthreads();

    // ---- exact-f32 skinny GEMM via V_WMMA_F32_16X16X4_F32, two chains ----
    // A fragment (16x4 f32): lanes 0-15 hold row=lane16, K = {k, k+1};
    //                        lanes 16-31 hold row=lane16, K = {k+2, k+3}.
    // B fragment (4x16 f32): lane holds column n=lane16, same K split.
    //   column 0/1 = weight vectors, columns 2-15 forced to zero (masked).
    v8f accV = {0.f, 0.f, 0.f, 0.f, 0.f, 0.f, 0.f, 0.f};
    v8f accE = {0.f, 0.f, 0.f, 0.f, 0.f, 0.f, 0.f, 0.f};

    const float bmask = (lane16 < 2) ? 1.0f : 0.0f;
    const int   sel   = lane16 & 1;
    const float* vrow = vbuf + lane16 * LDS_STRIDE + 2 * laneHalf;
    const float* erow = ebuf + lane16 * LDS_STRIDE + 2 * laneHalf;
    const float* wV   = wlds + sel * DIM       + 2 * laneHalf;   // chain V: w_ev (n=0), w_ee (n=1)
    const float* wE   = wlds + (2 + sel) * DIM + 2 * laneHalf;   // chain E: w_vv (n=0), w_ve (n=1)

    for (int k = 0; k < DIM; k += 4) {
        const v2f aV = *(const v2f*)(vrow + k);
        const v2f aE = *(const v2f*)(erow + k);
        v2f bV = *(const v2f*)(wV + k);
        v2f bE = *(const v2f*)(wE + k);
        bV.x *= bmask; bV.y *= bmask;
        bE.x *= bmask; bE.y *= bmask;
        accV = __builtin_amdgcn_wmma_f32_16x16x4_f32(false, aV, false, bV, (short)0, accV, false, false);
        accE = __builtin_amdgcn_wmma_f32_16x16x4_f32(false, aE, false, bE, (short)0, accE, false, false);
    }

    // ---- writer lanes (columns 0,1) publish per-row scalars ----
    // D layout: VGPR j -> row j (lanes 0-15) / row j+8 (lanes 16-31); lane%16 = column.
    if (lane16 < 2) {
        const int mb = laneHalf * 8;
        #pragma unroll
        for (int j = 0; j < 8; ++j) {
            sbuf[(mb + j) * 4 + sel]     = accV[j];   // slots 0,1: v.w_ev, v.w_ee
            sbuf[(mb + j) * 4 + 2 + sel] = accE[j];   // slots 2,3: e.w_vv, e.w_ve
        }
    }
    __syncthreads();

    // ---- epilogue: coalesced b128 LDS reads + b128 global stores ----
    for (int i = lane; i < ROWS_PER_WAVE * (DIM / 4); i += 32) {
        const int row = i >> 6;
        const int c4  = i & 63;
        const float4 vv = *(const float4*)(vbuf + row * LDS_STRIDE + c4 * 4);
        const float4 ee = *(const float4*)(ebuf + row * LDS_STRIDE + c4 * 4);
        const float4 s  = *(const float4*)(sbuf + row * 4);   // {v.w_ev, v.w_ee, e.w_vv, e.w_ve}
        const float4 bv = *(const float4*)(bias_v + c4 * 4);
        const float4 be = *(const float4*)(bias_e + c4 * 4);

        float4 ov, oe;
        // v_out = v*(e.w_vv) + e*(v.w_ev) + bias_v
        ov.x = vv.x * s.z + ee.x * s.x + bv.x;
        ov.y = vv.y * s.z + ee.y * s.x + bv.y;
        ov.z = vv.z * s.z + ee.z * s.x + bv.z;
        ov.w = vv.w * s.z + ee.w * s.x + bv.w;
        // e_out = v*(e.w_ve) + e*(v.w_ee) + bias_e
        oe.x = vv.x * s.w + ee.x * s.y + be.x;
        oe.y = vv.y * s.w + ee.y * s.y + be.y;
        oe.z = vv.z * s.w + ee.z * s.y + be.z;
        oe.w = vv.w * s.w + ee.w * s.y + be.w;

        const long off = (rowBase + row) * DIM + c4 * 4;
        *(float4*)(out_v + off) = ov;
        *(float4*)(out_e + off) = oe;
    }
}

extern "C" void kernel_launch(void* const* d_in, const int* in_sizes, int n_in,
                              void* d_out, int out_size, void* d_ws, size_t ws_size,
                              hipStream_t stream) {
    (void)n_in; (void)out_size; (void)d_ws; (void)ws_size;
    const float* v      = (const float*)d_in[0];
    const float* e      = (const float*)d_in[1];
    const float* w_vv   = (const float*)d_in[2];
    const float* w_ev   = (const float*)d_in[3];
    const float* w_ve   = (const float*)d_in[4];
    const float* w_ee   = (const float*)d_in[5];
    const float* bias_v = (const float*)d_in[6];
    const float* bias_e = (const float*)d_in[7];

    const int Bn = in_sizes[0] / DIM;                      // 131072
    float* out_v = (float*)d_out;
    float* out_e = out_v + (size_t)Bn * DIM;

    const size_t lds_bytes = LDS_TOTAL_FLOATS * sizeof(float);  // 138240 B (< 320 KB/WGP)
    // Dynamic LDS above 64 KB needs the opt-in attribute (no-op repeat calls are deterministic).
    hipFuncSetAttribute((const void*)cross_compress_kernel,
                        hipFuncAttributeMaxDynamicSharedMemorySize, (int)lds_bytes);

    dim3 grid(Bn / ROWS_PER_BLOCK);   // 2048
    dim3 block(128);                  // 4 waves (wave32)
    hipLaunchKernelGGL(cross_compress_kernel, grid, block, lds_bytes, stream,
                       v, e, w_vv, w_ev, w_ve, w_ee, bias_v, bias_e, out_v, out_e);
}